// GIN0WithJK_2611340116521
// MI455X (gfx1250) — compile-verified
//
#include <hip/hip_runtime.h>
#include <hip/hip_bf16.h>

#define GN 50000
#define GE 800000
#define GH 64
#define GL 4
#define GG 500
#define GC 10

typedef __attribute__((ext_vector_type(2)))  float  v2f;
typedef __attribute__((ext_vector_type(8)))  float  v8f;

// ---------------------------------------------------------------------------
// BN folding: out = relu(acc * s + c), s = g*rsqrt(v+eps), c = (b - m)*s + beta
// ---------------------------------------------------------------------------
__global__ void bn_precompute(const float* __restrict__ g1, const float* __restrict__ be1,
                              const float* __restrict__ m1, const float* __restrict__ v1,
                              const float* __restrict__ b1,
                              const float* __restrict__ g2, const float* __restrict__ be2,
                              const float* __restrict__ m2, const float* __restrict__ v2,
                              const float* __restrict__ b2,
                              float* __restrict__ s1, float* __restrict__ c1,
                              float* __restrict__ s2, float* __restrict__ c2) {
    int i = blockIdx.x * blockDim.x + threadIdx.x;   // 0 .. L*H-1 (256)
    if (i >= GL * GH) return;
    float sa = g1[i] * rsqrtf(v1[i] + 1e-5f);
    s1[i] = sa;
    c1[i] = (b1[i] - m1[i]) * sa + be1[i];
    float sb = g2[i] * rsqrtf(v2[i] + 1e-5f);
    s2[i] = sb;
    c2[i] = (b2[i] - m2[i]) * sb + be2[i];
}

// ---------------------------------------------------------------------------
// Edge scatter-add: agg[dst] += h[src].  16 lanes per edge, float4 per lane.
// ---------------------------------------------------------------------------
__global__ void edge_scatter(const float* __restrict__ h, long hstride,
                             const int* __restrict__ src, const int* __restrict__ dst,
                             float* __restrict__ agg, int nEdges) {
    int t = blockIdx.x * blockDim.x + threadIdx.x;
    int e = t >> 4;
    if (e >= nEdges) return;
    int c = (t & 15) << 2;
    int s = src[e], d = dst[e];
    const float4 v = *(const float4*)(h + (long)s * hstride + c);
    float* ap = agg + (long)d * GH + c;
    atomicAdd(ap + 0, v.x);
    atomicAdd(ap + 1, v.y);
    atomicAdd(ap + 2, v.z);
    atomicAdd(ap + 3, v.w);
}

// ---------------------------------------------------------------------------
// Fused GEMM: Out[n][j] = relu( (A[n] (+Aadd[n])) @ W  * scale[j] + bias[j] )
// A: nrows x 64 (row stride astride), W: 64x64 row-major [k][n].
// Block = 128 threads = 4 waves; each wave does 16 rows x 64 cols via
// V_WMMA_F32_16X16X4_F32.  HAS_ADD is a template flag (no divergent branch),
// all 16 A fragments are preloaded (b64 loads in flight), and W is stored
// pair-packed in LDS so every B fragment is one ds_load_b64.
// ---------------------------------------------------------------------------
template <bool HAS_ADD>
__global__ void __launch_bounds__(128)
gin_gemm64(const float* __restrict__ A, long astride,
           const float* __restrict__ Aadd, long addstride,
           const float* __restrict__ W,
           const float* __restrict__ scale, const float* __restrict__ bias,
           float* __restrict__ Out, long ostride, int nrows) {
    // pair-packed weights: sWp[t*64 + n] = { W[2t][n], W[2t+1][n] }
    __shared__ float2 sWp[(GH / 2) * GH];
    __shared__ float sS[GH], sB[GH];
    int tid = threadIdx.x;
    for (int i = tid; i < (GH / 2) * GH; i += 128) {
        int t = i >> 6, n = i & 63;
        sWp[i] = make_float2(W[(2 * t) * GH + n], W[(2 * t + 1) * GH + n]);
    }
    if (tid < GH) { sS[tid] = scale[tid]; sB[tid] = bias[tid]; }
    __syncthreads();

    const int wave = tid >> 5;
    const int lane = tid & 31;
    const int mlane = lane & 15;
    const int sel = lane >> 4;          // 0: lanes 0-15, 1: lanes 16-31
    const int koff = sel * 2;           // K offset within each K=4 step
    const int rowBase = blockIdx.x * 64 + wave * 16;

    int arow = rowBase + mlane;
    if (arow >= nrows) arow = nrows - 1;          // clamp loads; stores guarded
    const float* aptr = A + (long)arow * astride + koff;
    const float* adp  = Aadd + (HAS_ADD ? (long)arow * addstride + koff : 0);

    // Preload all A fragments: 16 (or 32) global b64 loads in flight.
    float2 areg[16];
#pragma unroll
    for (int i = 0; i < 16; ++i)
        areg[i] = *(const float2*)(aptr + 4 * i);
    if (HAS_ADD) {
#pragma unroll
        for (int i = 0; i < 16; ++i) {
            float2 d = *(const float2*)(adp + 4 * i);
            areg[i].x += d.x;
            areg[i].y += d.y;
        }
    }

    v8f acc0 = {}, acc1 = {}, acc2 = {}, acc3 = {};
#pragma unroll
    for (int i = 0; i < 16; ++i) {       // k0 = 4*i
        v2f a;
        a.x = areg[i].x;
        a.y = areg[i].y;
        // pair index t = k0/2 + sel
        const float2* bp = &sWp[(2 * i + sel) * GH];
        float2 f0 = bp[mlane];
        float2 f1 = bp[mlane + 16];
        float2 f2 = bp[mlane + 32];
        float2 f3 = bp[mlane + 48];
        v2f b0; b0.x = f0.x; b0.y = f0.y;
        v2f b1; b1.x = f1.x; b1.y = f1.y;
        v2f b2; b2.x = f2.x; b2.y = f2.y;
        v2f b3; b3.x = f3.x; b3.y = f3.y;
        acc0 = __builtin_amdgcn_wmma_f32_16x16x4_f32(false, a, false, b0, (short)0, acc0, false, false);
        acc1 = __builtin_amdgcn_wmma_f32_16x16x4_f32(false, a, false, b1, (short)0, acc1, false, false);
        acc2 = __builtin_amdgcn_wmma_f32_16x16x4_f32(false, a, false, b2, (short)0, acc2, false, false);
        acc3 = __builtin_amdgcn_wmma_f32_16x16x4_f32(false, a, false, b3, (short)0, acc3, false, false);
    }

    // C/D layout: VGPR r -> M = r (lanes 0-15) or M = 8+r (lanes 16-31), N = lane&15
    const int rbase = rowBase + (sel << 3);
    const int col = mlane;
    const float s0 = sS[col],      cc0 = sB[col];
    const float s1 = sS[col + 16], cc1 = sB[col + 16];
    const float s2 = sS[col + 32], cc2 = sB[col + 32];
    const float s3 = sS[col + 48], cc3 = sB[col + 48];
#pragma unroll
    for (int r = 0; r < 8; ++r) {
        int row = rbase + r;
        if (row < nrows) {
            float* o = Out + (long)row * ostride;
            o[col]      = fmaxf(acc0[r] * s0 + cc0, 0.f);
            o[col + 16] = fmaxf(acc1[r] * s1 + cc1, 0.f);
            o[col + 32] = fmaxf(acc2[r] * s2 + cc2, 0.f);
            o[col + 48] = fmaxf(acc3[r] * s3 + cc3, 0.f);
        }
    }
}

// ---------------------------------------------------------------------------
// Sorted-segment pooled sum: pooled[batch[n]][c] += xcat[n][c].
// 256 threads (one per feature), run-length accumulate over a node strip,
// flush one atomic per graph transition.
// ---------------------------------------------------------------------------
#define POOL_STRIP 256
__global__ void pool_segsum(const float* __restrict__ xcat, const int* __restrict__ batch,
                            float* __restrict__ pooled, int nrows) {
    int c = threadIdx.x;                         // 0..255
    int n0 = blockIdx.x * POOL_STRIP;
    int n1 = n0 + POOL_STRIP;
    if (n1 > nrows) n1 = nrows;
    float sum = 0.f;
    int cur = -1;
    for (int n = n0; n < n1; ++n) {
        int b = batch[n];
        if (b != cur) {
            if (cur >= 0) atomicAdd(&pooled[(long)cur * (GL * GH) + c], sum);
            cur = b;
            sum = 0.f;
        }
        sum += xcat[(long)n * (GL * GH) + c];
    }
    if (cur >= 0) atomicAdd(&pooled[(long)cur * (GL * GH) + c], sum);
}

// ---------------------------------------------------------------------------
// Final MLP: out = relu(pooled @ lin1_W + b1) @ lin2_W + b2.  One block/graph.
// ---------------------------------------------------------------------------
__global__ void final_mlp(const float* __restrict__ pooled,
                          const float* __restrict__ W1, const float* __restrict__ b1,
                          const float* __restrict__ W2, const float* __restrict__ b2,
                          float* __restrict__ out) {
    __shared__ float sh[GH];
    int g = blockIdx.x, j = threadIdx.x;         // 64 threads
    const float* p = pooled + (long)g * (GL * GH);
    float acc = b1[j];
    for (int k = 0; k < GL * GH; ++k) acc += p[k] * W1[k * GH + j];
    sh[j] = fmaxf(acc, 0.f);
    __syncthreads();
    if (j < GC) {
        float o = b2[j];
        for (int k = 0; k < GH; ++k) o += sh[k] * W2[k * GC + j];
        out[(long)g * GC + j] = o;
    }
}

// ---------------------------------------------------------------------------
extern "C" void kernel_launch(void* const* d_in, const int* in_sizes, int n_in,
                              void* d_out, int out_size, void* d_ws, size_t ws_size,
                              hipStream_t stream) {
    const float* x       = (const float*)d_in[0];
    const int*   ei      = (const int*)d_in[1];
    const int*   batch   = (const int*)d_in[2];
    const float* conv_W1 = (const float*)d_in[3];
    const float* conv_b1 = (const float*)d_in[4];
    const float* bn1_g   = (const float*)d_in[5];
    const float* bn1_b   = (const float*)d_in[6];
    const float* bn1_m   = (const float*)d_in[7];
    const float* bn1_v   = (const float*)d_in[8];
    const float* conv_W2 = (const float*)d_in[9];
    const float* conv_b2 = (const float*)d_in[10];
    const float* bn2_g   = (const float*)d_in[11];
    const float* bn2_b   = (const float*)d_in[12];
    const float* bn2_m   = (const float*)d_in[13];
    const float* bn2_v   = (const float*)d_in[14];
    const float* lin1_W  = (const float*)d_in[15];
    const float* lin1_b  = (const float*)d_in[16];
    const float* lin2_W  = (const float*)d_in[17];
    const float* lin2_b  = (const float*)d_in[18];
    float* out = (float*)d_out;

    const int* src = ei;
    const int* dst = ei + GE;

    // workspace carve-up (floats)
    float* ws     = (float*)d_ws;
    float* agg    = ws;                              // N*H
    float* tmp    = agg + (size_t)GN * GH;           // N*H
    float* xs     = tmp + (size_t)GN * GH;           // N*L*H (JK concat buffer)
    float* pooled = xs + (size_t)GN * GL * GH;       // G*L*H
    float* sc     = pooled + (size_t)GG * GL * GH;   // 4*L*H
    float* s1 = sc, *c1 = sc + GL * GH, *s2 = sc + 2 * GL * GH, *c2 = sc + 3 * GL * GH;

    bn_precompute<<<1, 256, 0, stream>>>(bn1_g, bn1_b, bn1_m, bn1_v, conv_b1,
                                         bn2_g, bn2_b, bn2_m, bn2_v, conv_b2,
                                         s1, c1, s2, c2);
    hipMemsetAsync(pooled, 0, (size_t)GG * GL * GH * sizeof(float), stream);

    const int edgeBlocks = (GE * 16 + 255) / 256;
    const int gemmBlocks = (GN + 63) / 64;

    for (int l = 0; l < GL; ++l) {
        const float* h    = (l == 0) ? x : (xs + (size_t)(l - 1) * GH);
        long         hstr = (l == 0) ? GH : (GL * GH);

        hipMemsetAsync(agg, 0, (size_t)GN * GH * sizeof(float), stream);
        edge_scatter<<<edgeBlocks, 256, 0, stream>>>(h, hstr, src, dst, agg, GE);

        // z = relu(bn1((h+agg) @ W1 + b1))
        gin_gemm64<true><<<gemmBlocks, 128, 0, stream>>>(
            h, hstr, agg, GH,
            conv_W1 + (size_t)l * GH * GH,
            s1 + l * GH, c1 + l * GH,
            tmp, GH, GN);
        // z = relu(bn2(z @ W2 + b2)) -> written into JK slot l (stride L*H)
        gin_gemm64<false><<<gemmBlocks, 128, 0, stream>>>(
            tmp, GH, nullptr, 0,
            conv_W2 + (size_t)l * GH * GH,
            s2 + l * GH, c2 + l * GH,
            xs + (size_t)l * GH, GL * GH, GN);
    }

    pool_segsum<<<(GN + POOL_STRIP - 1) / POOL_STRIP, GL * GH, 0, stream>>>(xs, batch, pooled, GN);
    final_mlp<<<GG, GH, 0, stream>>>(pooled, lin1_W, lin1_b, lin2_W, lin2_b, out);
}